// DecoderStandardBP_20813411516877
// MI455X (gfx1250) — compile-verified
//
#include <hip/hip_runtime.h>
#include <math.h>

// Sparse-edge belief propagation decoder for MI455X (gfx1250).
//
// The Tanner graph has only ~3.1K edges, so dense [B,C,V] emulation (128MB
// tensors x5 iters) is ~1000x wasted bandwidth. Instead: one workgroup per
// batch element (64 WGs x 512 threads = 16 wave32s), one thread per check
// row, all decoder state LDS-resident (~104KB/WG out of 320KB/WGP).
//
// CDNA5-specific paths:
//   * global_load_async_to_lds_b32 (+ s_wait_asynccnt) DMAs the channel
//     samples into LDS while the adjacency is acquired
//   * ds_add_f32 LDS atomics for the per-variable scatter-add of messages
// WMMA is deliberately NOT used: S[v] = sum_c c2v[c,v] is a sparse
// scatter-add with no dense matmul substructure.
//
// Adjacency is built ONCE by a wave-per-row ballot-compaction kernel into
// d_ws (all 32 row loads issued before one wait -> no serialized round
// trips); if ws_size is too small, the decode kernel falls back to an
// in-kernel scan with register-batched loads (8x float4 per wait).

#define N_VAR  1024
#define N_CHK  512
#define N_INFO 512
#define N_ITER 5
#define BATCH  64
#define MAXDEG 32            // Binomial(1024, 6/1024): P(deg>24) ~ 0; 32 is safe
#define EPS_T  1e-7f
#define CLIP_T (1.0f - 1e-6f)

#define ADJ_DEG_BYTES (N_CHK * (int)sizeof(int))
#define ADJ_IDX_BYTES (N_CHK * MAXDEG * (int)sizeof(unsigned short))
#define ADJ_BYTES     (ADJ_DEG_BYTES + ADJ_IDX_BYTES)

// ---------------------------------------------------------------------------
// Adjacency build: one wave32 per check row. All 32 coalesced loads of the
// row are issued before consumption (single s_wait), then ballot-compacted
// in ascending column order (group-ascending, lane-ascending).
// ---------------------------------------------------------------------------
__global__ __launch_bounds__(256)
void build_adj_kernel(const float* __restrict__ H,
                      int* __restrict__ deg_out,
                      unsigned short* __restrict__ idx_out)
{
    const int gtid = blockIdx.x * blockDim.x + threadIdx.x;
    const int row  = gtid >> 5;        // wave id == check row
    const int lane = gtid & 31;
    if (row >= N_CHK) return;

    const float* __restrict__ r = H + (size_t)row * N_VAR;

    // Issue all 32 coalesced loads up front (independent -> pipelined).
    float vals[32];
    #pragma unroll
    for (int g = 0; g < 32; ++g) vals[g] = r[g * 32 + lane];

    int deg = 0;
    #pragma unroll
    for (int g = 0; g < 32; ++g) {
        bool on = vals[g] > 0.5f;
        unsigned m = (unsigned)__ballot(on);
        if (on) {
            int pos = deg + __popc(m & ((1u << lane) - 1u));
            if (pos < MAXDEG)
                idx_out[row * MAXDEG + pos] = (unsigned short)(g * 32 + lane);
        }
        deg += __popc(m);
    }
    if (lane == 0) deg_out[row] = (deg < MAXDEG) ? deg : MAXDEG;
}

// ---------------------------------------------------------------------------
// Decoder: one workgroup per batch element.
// ---------------------------------------------------------------------------
__global__ __launch_bounds__(N_CHK)
void bp_decode_kernel(const float* __restrict__ inputs,   // [B, V]
                      const float* __restrict__ H,        // [C, V] 0/1 mask
                      const float* __restrict__ sigma2,   // [1]
                      float* __restrict__ out,            // [B, N_INFO]
                      const int* __restrict__ deg_ws,     // [C]   (if use_ws)
                      const unsigned short* __restrict__ idx_ws, // [C,MAXDEG]
                      int use_ws)
{
    __shared__ float          llr_s[N_VAR];              // 4 KB
    __shared__ float          S[N_VAR];                  // 4 KB (sum_c c2v per variable)
    __shared__ unsigned short idx_s[N_CHK * MAXDEG];     // 32 KB (per-row var indices)
    __shared__ float          c2v_s[N_CHK * MAXDEG];     // 64 KB (per-edge messages)

    const int tid = threadIdx.x;   // 0..511: this thread owns check row `tid`
    const int b   = blockIdx.x;    // batch element

    // ---- 1) Kick off async DMA of this batch's 1024 channel samples to LDS.
    //         Each thread pulls 2 dwords; tracked by ASYNCcnt.
    {
        unsigned lds0 = (unsigned)(size_t)&llr_s[tid];
        unsigned lds1 = (unsigned)(size_t)&llr_s[tid + N_CHK];
        unsigned off0 = (unsigned)((b * N_VAR + tid) * sizeof(float));
        unsigned off1 = off0 + (unsigned)(N_CHK * sizeof(float));
        unsigned long long gbase = (unsigned long long)inputs;
        asm volatile("global_load_async_to_lds_b32 %0, %1, %2"
                     :: "v"(lds0), "v"(off0), "s"(gbase) : "memory");
        asm volatile("global_load_async_to_lds_b32 %0, %1, %2"
                     :: "v"(lds1), "v"(off1), "s"(gbase) : "memory");
    }

    // ---- 2) Acquire this row's adjacency while the DMA is in flight.
    const int rb = tid * MAXDEG;
    int deg;
    if (use_ws) {
        // Precomputed by build_adj_kernel: compact 34KB, L2-resident.
        deg = deg_ws[tid];
        for (int j = 0; j < deg; ++j) idx_s[rb + j] = idx_ws[rb + j];
    } else {
        // Fallback: scan my H row with register-batched loads (8x float4
        // issued per wait -> 32 round trips instead of 256).
        deg = 0;
        const float4* __restrict__ Hrow = (const float4*)(H + (size_t)tid * N_VAR);
        for (int blk = 0; blk < N_VAR / (4 * 8); ++blk) {
            float4 buf[8];
            #pragma unroll
            for (int u = 0; u < 8; ++u) buf[u] = Hrow[blk * 8 + u];
            #pragma unroll
            for (int u = 0; u < 8; ++u) {
                float4 h  = buf[u];
                int base  = (blk * 8 + u) * 4;
                if (h.x > 0.5f) { if (deg < MAXDEG) idx_s[rb + deg] = (unsigned short)(base + 0); ++deg; }
                if (h.y > 0.5f) { if (deg < MAXDEG) idx_s[rb + deg] = (unsigned short)(base + 1); ++deg; }
                if (h.z > 0.5f) { if (deg < MAXDEG) idx_s[rb + deg] = (unsigned short)(base + 2); ++deg; }
                if (h.w > 0.5f) { if (deg < MAXDEG) idx_s[rb + deg] = (unsigned short)(base + 3); ++deg; }
            }
        }
        if (deg > MAXDEG) deg = MAXDEG;
    }

    // zero my c2v row and my slice of S
    #pragma unroll
    for (int j = 0; j < MAXDEG; ++j) c2v_s[rb + j] = 0.0f;
    S[tid] = 0.0f;
    S[tid + N_CHK] = 0.0f;

    // ---- 3) Wait for my async DMA, sync the workgroup, scale LLRs in place.
    asm volatile("s_wait_asynccnt 0" ::: "memory");
    __syncthreads();
    const float scale = -4.0f / sigma2[0];
    llr_s[tid]         *= scale;
    llr_s[tid + N_CHK] *= scale;
    __syncthreads();

    // ---- 4) 5 BP iterations, fully LDS-resident.
    for (int it = 0; it < N_ITER; ++it) {
        // Phase 1: per-edge tanh of extrinsic v->c messages; full row product
        // (same full-product-then-divide leave-one-out as the reference;
        //  off-edge factors are exact 1.0s so ordering matches bitwise).
        float prod = 1.0f;
        for (int j = 0; j < deg; ++j) {
            int   v   = idx_s[rb + j];
            float v2c = llr_s[v] + S[v] - c2v_s[rb + j];
            float t   = tanhf(0.5f * v2c);
            float sgn = (t >= 0.0f) ? 1.0f : -1.0f;
            float a   = fminf(fmaxf(fabsf(t), EPS_T), CLIP_T);
            t = sgn * a;
            c2v_s[rb + j] = t;      // stash t (old c2v already consumed)
            prod *= t;
        }
        __syncthreads();            // everyone done reading old S
        S[tid] = 0.0f;
        S[tid + N_CHK] = 0.0f;
        __syncthreads();            // S cleared before accumulation

        // Phase 2: leave-one-out c->v messages; scatter-add into S
        // (ds_add_f32 LDS atomics).
        for (int j = 0; j < deg; ++j) {
            int   v   = idx_s[rb + j];
            float t   = c2v_s[rb + j];
            float ext = fminf(fmaxf(prod / t, -CLIP_T), CLIP_T);
            float nc  = logf((1.0f + ext) / (1.0f - ext));   // 2*atanh(ext)
            c2v_s[rb + j] = nc;
            atomicAdd(&S[v], nc);
        }
        __syncthreads();            // new S visible for next iteration
    }

    // ---- 5) Posterior LLR + sigmoid on the N_INFO information bits.
    //         sigmoid(-x) = 1 / (1 + e^x).   (tid in [0,512) == info bit id)
    float total = llr_s[tid] + S[tid];
    out[(size_t)b * N_INFO + tid] = 1.0f / (1.0f + expf(total));
}

extern "C" void kernel_launch(void* const* d_in, const int* in_sizes, int n_in,
                              void* d_out, int out_size, void* d_ws, size_t ws_size,
                              hipStream_t stream) {
    (void)in_sizes; (void)n_in; (void)out_size;
    const float* inputs = (const float*)d_in[0];   // [64, 1024] f32
    const float* H      = (const float*)d_in[1];   // [512, 1024] f32 0/1
    const float* sigma2 = (const float*)d_in[2];   // [1] f32
    float*       out    = (float*)d_out;           // [64, 512] f32

    int*            deg_ws = (int*)d_ws;
    unsigned short* idx_ws = (unsigned short*)((char*)d_ws + ADJ_DEG_BYTES);
    const int use_ws = (d_ws != nullptr && ws_size >= (size_t)ADJ_BYTES) ? 1 : 0;

    if (use_ws) {
        // 512 waves, one per check row; 64 blocks x 8 wave32s.
        build_adj_kernel<<<dim3((N_CHK * 32) / 256), dim3(256), 0, stream>>>(
            H, deg_ws, idx_ws);
    }
    bp_decode_kernel<<<dim3(BATCH), dim3(N_CHK), 0, stream>>>(
        inputs, H, sigma2, out, deg_ws, idx_ws, use_ws);
}